// PointHead_4423816315274
// MI455X (gfx1250) — compile-verified
//
#include <hip/hip_runtime.h>
#include <hip/hip_bf16.h>

typedef float v2f __attribute__((ext_vector_type(2)));
typedef float v8f __attribute__((ext_vector_type(8)));

#define BATCH     8
#define NOG       192
#define NCOV      16
#define NIMP      48
#define NPTS      64
#define CFINE     512
#define HWF       256
#define HWC       64
#define KDIM      514   // 2 coarse + 512 fine channels
#define KPAD      516   // rounded up to multiple of 4 for 16x16x4 WMMA

// ---------------------------------------------------------------------------
// Kernel 1: uncertainty at the 192 over-generated points per batch.
// mask_sorted = per-pixel (max,min) of the two `out` channels; bilinear sample
// with zeros padding; uncertainty = sampled(min) - sampled(max).
// ---------------------------------------------------------------------------
__global__ void uncert_kernel(const float* __restrict__ outmap,
                              const float* __restrict__ og,
                              float* __restrict__ u) {
    int g = blockIdx.x * blockDim.x + threadIdx.x;
    if (g >= BATCH * NOG) return;
    int b = g / NOG, i = g - b * NOG;

    float px = og[(b * NOG + i) * 2 + 0];
    float py = og[(b * NOG + i) * 2 + 1];
    // grid_sample align_corners=False: x = px*W - 0.5
    float xf = px * (float)HWC - 0.5f;
    float yf = py * (float)HWC - 0.5f;
    float x0 = floorf(xf), y0 = floorf(yf);
    float wx = xf - x0,    wy = yf - y0;
    int ix = (int)x0, iy = (int)y0;

    const float* c0 = outmap + (size_t)(b * 2 + 0) * HWC * HWC;
    const float* c1 = outmap + (size_t)(b * 2 + 1) * HWC * HWC;

    float wt[4] = {(1.f - wy) * (1.f - wx), (1.f - wy) * wx,
                   wy * (1.f - wx),         wy * wx};
    int tx[4] = {ix, ix + 1, ix, ix + 1};
    int ty[4] = {iy, iy, iy + 1, iy + 1};

    float smax = 0.f, smin = 0.f;
#pragma unroll
    for (int t = 0; t < 4; ++t) {
        bool valid = (tx[t] >= 0) && (tx[t] < HWC) && (ty[t] >= 0) && (ty[t] < HWC);
        int cx = min(max(tx[t], 0), HWC - 1);
        int cy = min(max(ty[t], 0), HWC - 1);
        float a = valid ? c0[cy * HWC + cx] : 0.f;
        float v = valid ? c1[cy * HWC + cx] : 0.f;
        smax += wt[t] * fmaxf(a, v);
        smin += wt[t] * fminf(a, v);
    }
    u[b * NOG + i] = smin - smax;   // -(og0 - og1)
}

// ---------------------------------------------------------------------------
// Kernel 2: stable top-48 per batch via O(192^2) rank counting (matches
// jax.lax.top_k: descending values, ties prefer lower index), then append
// the 16 coverage points. Writes points to workspace AND to output slot 2.
// ---------------------------------------------------------------------------
__global__ void select_kernel(const float* __restrict__ u,
                              const float* __restrict__ og,
                              const float* __restrict__ cov,
                              float* __restrict__ pts_ws,
                              float* __restrict__ pts_out) {
    __shared__ float ul[NOG];
    int b = blockIdx.x;
    int i = threadIdx.x;              // 0..191
    ul[i] = u[b * NOG + i];
    __syncthreads();

    float ui = ul[i];
    int rank = 0;
    for (int j = 0; j < NOG; ++j) {
        float uj = ul[j];
        rank += (uj > ui) || (uj == ui && j < i);
    }
    if (rank < NIMP) {
        float px = og[(b * NOG + i) * 2 + 0];
        float py = og[(b * NOG + i) * 2 + 1];
        int o = (b * NPTS + rank) * 2;
        pts_ws[o + 0] = px;  pts_ws[o + 1] = py;
        pts_out[o + 0] = px; pts_out[o + 1] = py;
    }
    if (i < NCOV) {
        float px = cov[(b * NCOV + i) * 2 + 0];
        float py = cov[(b * NCOV + i) * 2 + 1];
        int o = (b * NPTS + NIMP + i) * 2;
        pts_ws[o + 0] = px;  pts_ws[o + 1] = py;
        pts_out[o + 0] = px; pts_out[o + 1] = py;
    }
}

// ---------------------------------------------------------------------------
// Kernel 3: one workgroup per (batch, 16-point tile). 256 threads gather the
// 514x16 feature tile (coarse from `out`, fine from `res2`) into LDS with
// zero-padded rows 514/515, and stage the zero-padded 16x516 weight matrix
// into LDS in transposed [k][m] layout (padding resolved at fill time so the
// WMMA loop has NO conditional loads / EXEC churn). Wave 0 then contracts
// with V_WMMA_F32_16X16X4_F32, K looped in steps of 4: two ds_load_2addr
// pairs + one wmma per step.
// C/D layout: reg r, lanes 0-15 => row r  => rows 0,1 = real outputs.
// ---------------------------------------------------------------------------
__global__ void __launch_bounds__(256)
render_kernel(const float* __restrict__ outmap,
              const float* __restrict__ res2,
              const float* __restrict__ wmat,
              const float* __restrict__ bias,
              const float* __restrict__ pts,
              float* __restrict__ rend) {
    __shared__ float feat[KPAD * 16];   // B operand: [k][p], row-stride 16
    __shared__ float wpad[KPAD * 16];   // A operand: [k][m], zero-padded
    __shared__ float ppx[16], ppy[16];

    int b  = blockIdx.x >> 2;
    int n0 = (blockIdx.x & 3) * 16;
    int tid = threadIdx.x;

    if (tid < 16) {
        ppx[tid] = pts[(b * NPTS + n0 + tid) * 2 + 0];
        ppy[tid] = pts[(b * NPTS + n0 + tid) * 2 + 1];
    }

    // Stage zero-padded weights: wpad[k][m] = W[m][k] for m<2, k<514 else 0.
    // All padding conditionals resolved here, outside the WMMA loop.
    for (int idx = tid; idx < KPAD * 16; idx += 256) {
        int k = idx >> 4;
        int m = idx & 15;
        wpad[idx] = (m < 2 && k < KDIM) ? wmat[m * KDIM + k] : 0.f;
    }
    __syncthreads();

    int p = tid & 15;      // point within tile
    int g = tid >> 4;      // channel group 0..15
    float px = ppx[p], py = ppy[p];

    // --- fine (256x256) tap parameters, shared across this thread's channels
    {
        float xf = px * (float)HWF - 0.5f;
        float yf = py * (float)HWF - 0.5f;
        float x0 = floorf(xf), y0 = floorf(yf);
        float wx = xf - x0,    wy = yf - y0;
        int ix = (int)x0, iy = (int)y0;
        bool vx0 = (ix >= 0) && (ix < HWF);
        bool vx1 = (ix + 1 >= 0) && (ix + 1 < HWF);
        bool vy0 = (iy >= 0) && (iy < HWF);
        bool vy1 = (iy + 1 >= 0) && (iy + 1 < HWF);
        int cx0 = min(max(ix, 0), HWF - 1),  cx1 = min(max(ix + 1, 0), HWF - 1);
        int cy0 = min(max(iy, 0), HWF - 1),  cy1 = min(max(iy + 1, 0), HWF - 1);
        float w00 = (vy0 && vx0) ? (1.f - wy) * (1.f - wx) : 0.f;
        float w01 = (vy0 && vx1) ? (1.f - wy) * wx         : 0.f;
        float w10 = (vy1 && vx0) ? wy * (1.f - wx)         : 0.f;
        float w11 = (vy1 && vx1) ? wy * wx                 : 0.f;
        int i00 = cy0 * HWF + cx0, i01 = cy0 * HWF + cx1;
        int i10 = cy1 * HWF + cx0, i11 = cy1 * HWF + cx1;
#pragma unroll 4
        for (int ii = 0; ii < 32; ++ii) {
            int c = g + 16 * ii;                       // 0..511
            const float* base = res2 + ((size_t)(b * CFINE + c)) * (HWF * HWF);
            float v = w00 * base[i00] + w01 * base[i01]
                    + w10 * base[i10] + w11 * base[i11];
            feat[(2 + c) * 16 + p] = v;
        }
    }

    if (g == 0) {
        // coarse (64x64), channels 0..1 of raw `out`
        float xf = px * (float)HWC - 0.5f;
        float yf = py * (float)HWC - 0.5f;
        float x0 = floorf(xf), y0 = floorf(yf);
        float wx = xf - x0,    wy = yf - y0;
        int ix = (int)x0, iy = (int)y0;
        bool vx0 = (ix >= 0) && (ix < HWC);
        bool vx1 = (ix + 1 >= 0) && (ix + 1 < HWC);
        bool vy0 = (iy >= 0) && (iy < HWC);
        bool vy1 = (iy + 1 >= 0) && (iy + 1 < HWC);
        int cx0 = min(max(ix, 0), HWC - 1),  cx1 = min(max(ix + 1, 0), HWC - 1);
        int cy0 = min(max(iy, 0), HWC - 1),  cy1 = min(max(iy + 1, 0), HWC - 1);
        float w00 = (vy0 && vx0) ? (1.f - wy) * (1.f - wx) : 0.f;
        float w01 = (vy0 && vx1) ? (1.f - wy) * wx         : 0.f;
        float w10 = (vy1 && vx0) ? wy * (1.f - wx)         : 0.f;
        float w11 = (vy1 && vx1) ? wy * wx                 : 0.f;
#pragma unroll
        for (int ch = 0; ch < 2; ++ch) {
            const float* base = outmap + (size_t)(b * 2 + ch) * (HWC * HWC);
            float v = w00 * base[cy0 * HWC + cx0] + w01 * base[cy0 * HWC + cx1]
                    + w10 * base[cy1 * HWC + cx0] + w11 * base[cy1 * HWC + cx1];
            feat[ch * 16 + p] = v;
        }
    }
    if (g == 1) {          // zero K-padding rows of the B tile
        feat[514 * 16 + p] = 0.f;
        feat[515 * 16 + p] = 0.f;
    }
    __syncthreads();

    // --- wave 0 only: uniform branch, EXEC all-ones inside the wave -------
    if (tid < 32) {
        int lane = tid;
        int m  = lane & 15;          // A row (M) / B column (N)
        int kb = (lane >> 4) * 2;    // lane-group K offset: 0 or 2
        int kidx = kb * 16 + m;      // LDS element index at K=kb, col m

        v8f acc = {};
#pragma unroll 4
        for (int kk = 0; kk < KPAD; kk += 4) {
            int e0 = kk * 16 + kidx;         // element index of K=kk+kb
            v2f a, bb;
            a.x  = wpad[e0];
            a.y  = wpad[e0 + 16];
            bb.x = feat[e0];
            bb.y = feat[e0 + 16];
            acc = __builtin_amdgcn_wmma_f32_16x16x4_f32(
                false, a, false, bb, (short)0, acc, false, false);
        }
        // D rows 0 and 1 live in acc[0]/acc[1] of lanes 0..15
        if (lane < 16) {
            rend[b * (2 * NPTS) + 0 * NPTS + n0 + lane] = acc[0] + bias[0];
            rend[b * (2 * NPTS) + 1 * NPTS + n0 + lane] = acc[1] + bias[1];
        }
    }
}

// ---------------------------------------------------------------------------
extern "C" void kernel_launch(void* const* d_in, const int* in_sizes, int n_in,
                              void* d_out, int out_size, void* d_ws, size_t ws_size,
                              hipStream_t stream) {
    (void)in_sizes; (void)n_in; (void)out_size; (void)ws_size;
    // inputs: x(unused), res2, out, over_generation, coverage, w, b
    const float* res2 = (const float*)d_in[1];
    const float* outm = (const float*)d_in[2];
    const float* og   = (const float*)d_in[3];
    const float* cov  = (const float*)d_in[4];
    const float* wmat = (const float*)d_in[5];
    const float* bias = (const float*)d_in[6];

    float* rend    = (float*)d_out;                       // [8,2,64]
    float* pts_out = (float*)d_out + BATCH * 2 * NPTS;    // [8,64,2]

    float* u      = (float*)d_ws;                         // [8,192]
    float* pts_ws = u + BATCH * NOG;                      // [8,64,2]

    uncert_kernel<<<(BATCH * NOG + 255) / 256, 256, 0, stream>>>(outm, og, u);
    select_kernel<<<BATCH, NOG, 0, stream>>>(u, og, cov, pts_ws, pts_out);
    render_kernel<<<BATCH * 4, 256, 0, stream>>>(outm, res2, wmat, bias,
                                                 pts_ws, rend);
}